// NodeBlock_16449724745526
// MI455X (gfx1250) — compile-verified
//
#include <hip/hip_runtime.h>

#define NNODES  100000
#define NEDGES  1000000
#define DFEAT   64
#define LATENT  32
#define ODIM    64

typedef __attribute__((ext_vector_type(2))) float v2f;
typedef __attribute__((ext_vector_type(8))) float v8f;

#define WMMA_F32(a, b, c) \
    __builtin_amdgcn_wmma_f32_16x16x4_f32(false, (a), false, (b), (short)0, (c), false, false)

// ---------------------------------------------------------------- init ws
// float4 stores; n4 = number of float4 chunks (caller pads allocation usage)
__global__ __launch_bounds__(256) void init_ws_kernel(float4* __restrict__ ws,
                                                      long long n4) {
    long long i = (long long)blockIdx.x * blockDim.x + threadIdx.x;
    if (i < n4) ws[i] = make_float4(0.f, 0.f, 0.f, 0.f);
}

// ------------------------------------------------------- edge scatter-add
// one thread per (edge, 4-feature group): coalesced global_load_b128 of
// edge_attr (256 MB stream @ 23.3 TB/s floor), atomics land in the
// L2-resident 25.6 MB accumulator.
__global__ __launch_bounds__(256) void edge_scatter_kernel(
    const float4* __restrict__ edge_attr4,
    const int*    __restrict__ recv,
    float* __restrict__ agg,
    float* __restrict__ cnt) {
    long long gid = (long long)blockIdx.x * blockDim.x + threadIdx.x;
    if (gid >= (long long)NEDGES * (DFEAT / 4)) return;
    const int e = (int)(gid >> 4);        // 16 float4 chunks per edge
    const int q = (int)(gid & 15);        // chunk index -> features 4q..4q+3
    const int r = recv[e];
    const float4 v = edge_attr4[gid];
    float* dst = agg + (size_t)r * DFEAT + 4 * q;
    atomicAdd(dst + 0, v.x);
    atomicAdd(dst + 1, v.y);
    atomicAdd(dst + 2, v.z);
    atomicAdd(dst + 3, v.w);
    if (q == 0) atomicAdd(&cnt[r], 1.0f);
}

// ------------------------------------------------------- per-node MLP (WMMA)
// one wave32 per 16-node tile; 100000 = 6250*16 so EXEC is always all-ones.
__global__ __launch_bounds__(32) void node_mlp_kernel(
    const float* __restrict__ node_attr,
    const float* __restrict__ global_attr,
    const float* __restrict__ W1,
    const float* __restrict__ B1,
    const float* __restrict__ W2,
    const float* __restrict__ B2,
    const int*   __restrict__ ng_index,
    const float* __restrict__ agg,
    const float* __restrict__ cnt,
    float* __restrict__ out) {
    __shared__ float Hlds[16][33];   // 16x32 hidden tile, padded stride -> no bank conflicts

    const int lane = threadIdx.x;    // 0..31
    const int n    = lane & 15;      // A-row / B-col / D-col index
    const int hi   = lane >> 4;      // half-wave selects K pair (A/B) or row block (D)
    const int tile = blockIdx.x;
    const int node = tile * 16 + n;  // node whose feature row this lane feeds as A

    // per-lane source rows for the virtual concat [node | agg_mean | global]
    const float* xn = node_attr   + (size_t)node * DFEAT;
    const float* xa = agg         + (size_t)node * DFEAT;
    const float* xg = global_attr + (size_t)ng_index[node] * DFEAT;
    const float  invc = 1.0f / fmaxf(cnt[node], 1.0f);

    // ---------------- layer 1: [16x192] x [192x32] via 2 N-tiles * 48 K-steps
    v8f acc0 = {}; v8f acc1 = {};
#pragma unroll
    for (int ks = 0; ks < 48; ++ks) {
        const int k = ks * 4 + hi * 2;   // K column pair for this lane (A layout)
        v2f a;
        if (ks < 16)      { a.x = xn[k];              a.y = xn[k + 1]; }
        else if (ks < 32) { a.x = xa[k - 64] * invc;  a.y = xa[k - 63] * invc; }
        else              { a.x = xg[k - 128];        a.y = xg[k - 127]; }
        v2f b0, b1;        // B layout: lane holds col n, VGPR pair holds K pair
        b0.x = W1[(size_t)k * LATENT + n];
        b0.y = W1[(size_t)(k + 1) * LATENT + n];
        b1.x = W1[(size_t)k * LATENT + 16 + n];
        b1.y = W1[(size_t)(k + 1) * LATENT + 16 + n];
        acc0 = WMMA_F32(a, b0, acc0);
        acc1 = WMMA_F32(a, b1, acc1);
    }

    // bias + relu, transpose H through LDS (D layout: row = r + 8*hi, col = n)
#pragma unroll
    for (int r = 0; r < 8; ++r) {
        const int row = r + hi * 8;
        Hlds[row][n]      = fmaxf(acc0[r] + B1[n],      0.0f);
        Hlds[row][16 + n] = fmaxf(acc1[r] + B1[16 + n], 0.0f);
    }
    __syncthreads();

    // ---------------- layer 2: [16x32] x [32x64] via 4 N-tiles * 8 K-steps
    v8f o0 = {}, o1 = {}, o2 = {}, o3 = {};
#pragma unroll
    for (int ks = 0; ks < 8; ++ks) {
        const int k = ks * 4 + hi * 2;
        v2f a; a.x = Hlds[n][k]; a.y = Hlds[n][k + 1];
        v2f b0, b1, b2, b3;
        b0.x = W2[(size_t)k * ODIM + n];        b0.y = W2[(size_t)(k + 1) * ODIM + n];
        b1.x = W2[(size_t)k * ODIM + 16 + n];   b1.y = W2[(size_t)(k + 1) * ODIM + 16 + n];
        b2.x = W2[(size_t)k * ODIM + 32 + n];   b2.y = W2[(size_t)(k + 1) * ODIM + 32 + n];
        b3.x = W2[(size_t)k * ODIM + 48 + n];   b3.y = W2[(size_t)(k + 1) * ODIM + 48 + n];
        o0 = WMMA_F32(a, b0, o0);
        o1 = WMMA_F32(a, b1, o1);
        o2 = WMMA_F32(a, b2, o2);
        o3 = WMMA_F32(a, b3, o3);
    }

    // D layout -> coalesced stores of the 16x64 output tile
#pragma unroll
    for (int r = 0; r < 8; ++r) {
        float* orow = out + (size_t)(tile * 16 + r + hi * 8) * ODIM;
        orow[n]      = o0[r] + B2[n];
        orow[16 + n] = o1[r] + B2[16 + n];
        orow[32 + n] = o2[r] + B2[32 + n];
        orow[48 + n] = o3[r] + B2[48 + n];
    }
}

// ---------------------------------------------------------------- launcher
extern "C" void kernel_launch(void* const* d_in, const int* in_sizes, int n_in,
                              void* d_out, int out_size, void* d_ws, size_t ws_size,
                              hipStream_t stream) {
    const float* node_attr   = (const float*)d_in[0];
    const float* edge_attr   = (const float*)d_in[1];
    const float* global_attr = (const float*)d_in[2];
    const float* W1 = (const float*)d_in[3];
    const float* B1 = (const float*)d_in[4];
    const float* W2 = (const float*)d_in[5];
    const float* B2 = (const float*)d_in[6];
    const int* recv = (const int*)d_in[7];
    const int* ngi  = (const int*)d_in[8];
    float* out = (float*)d_out;

    float* agg = (float*)d_ws;                       // [NNODES * 64]
    float* cnt = agg + (size_t)NNODES * DFEAT;       // [NNODES]

    // zero agg + cnt (65 floats per node; 65 % 4 != 0 -> round chunks up,
    // NNODES*65 = 6,500,000 floats is float4-divisible: 1,625,000 chunks)
    const long long init_n4 = ((long long)NNODES * (DFEAT + 1) + 3) / 4;
    init_ws_kernel<<<(unsigned)((init_n4 + 255) / 256), 256, 0, stream>>>(
        (float4*)d_ws, init_n4);

    const long long scat_n = (long long)NEDGES * (DFEAT / 4);
    edge_scatter_kernel<<<(unsigned)((scat_n + 255) / 256), 256, 0, stream>>>(
        (const float4*)edge_attr, recv, agg, cnt);

    node_mlp_kernel<<<NNODES / 16, 32, 0, stream>>>(
        node_attr, global_attr, W1, B1, W2, B2, ngi, agg, cnt, out);
}